// GaussianDPMM_64055142252780
// MI455X (gfx1250) — compile-verified
//
#include <hip/hip_runtime.h>
#include <math.h>

typedef __attribute__((ext_vector_type(2))) float v2f;
typedef __attribute__((ext_vector_type(8))) float v8f;
typedef __attribute__((ext_vector_type(4))) unsigned int u32x4;
typedef __attribute__((ext_vector_type(8))) int i32x8;
typedef __attribute__((ext_vector_type(4))) int i32x4;

#define KC 128
#define DC 64
#define NC 131072
#define LOG_2PI_F 1.8378770664093453f
#define LN2_F 0.6931471805599453f

// workspace layout (float offsets)
#define WS_BPREP 0        // 128*128 swizzled B fragments (64 KB)
#define WS_CONST 16384    // 128 per-k logit constants
#define WS_KL    16512    // 1 (kl_beta + kl_ng)
#define WS_BSUM  16520    // 1024 per-block lse partial sums

__device__ __forceinline__ float digammaf_(float x) {
  // recurrence to x>=6 then asymptotic series; inputs here are always > 0
  float r = 0.f;
  while (x < 6.f) { r -= 1.f / x; x += 1.f; }
  float inv = 1.f / x, inv2 = inv * inv;
  return r + logf(x) - 0.5f * inv
       - inv2 * (1.f / 12.f - inv2 * (1.f / 120.f - inv2 * (1.f / 252.f)));
}

__device__ __forceinline__ float lnbetaf_(float a, float b) {
  return lgammaf(a) + lgammaf(b) - lgammaf(a + b);
}

// Swizzle inner-index i (0..127), column k (0..127) into WMMA B-fragment order:
// step kk = i>>2 ; within-step K-offset off = i&3 ;
// B frag (16x16x4 f32): VGPR j=off&1, lanes 0-15 hold K-off {0,1}, lanes 16-31 hold {2,3}
__device__ __forceinline__ int bprep_idx(int i, int k) {
  int kk = i >> 2, off = i & 3;
  int lane = ((off >> 1) << 4) | (k & 15);
  int t = k >> 4, j = off & 1;
  return ((kk * 8 + t) * 32 + lane) * 2 + j;
}

__global__ __launch_bounds__(128) void dpmm_prep(
    const float* __restrict__ nat_u, const float* __restrict__ nat_v,
    const float* __restrict__ nat_tau, const float* __restrict__ nat_c,
    const float* __restrict__ nat_n, const float* __restrict__ nat_B,
    const float* __restrict__ u0, const float* __restrict__ v0,
    const float* __restrict__ tau0, const float* __restrict__ c0,
    const float* __restrict__ n0, const float* __restrict__ B0,
    float* __restrict__ ws) {
  __shared__ float s_e1m[KC];
  __shared__ float s_kl[KC];
  int k = threadIdx.x;

  float u = nat_u[k] + 1.f, v = nat_v[k] + 1.f;
  float c = nat_c[k], n = nat_n[k];
  float dg_u = digammaf_(u), dg_v = digammaf_(v), dg_uv = digammaf_(u + v);
  s_e1m[k] = dg_v - dg_uv;
  __syncthreads();
  float cum = 0.f;
  for (int j = 0; j < k; ++j) cum += s_e1m[j];     // exclusive stick-breaking cumsum
  float e_log_pi = (dg_u - dg_uv) + cum;

  float a = 0.5f * n, a0 = 0.5f * n0[k];
  float dg_a = digammaf_(a);
  float lg_a = lgammaf(a), lg_a0 = lgammaf(a0);
  float c0k = c0[k];
  float invc = 1.f / c;

  float sum_loglam = 0.f, sum_wtau2 = 0.f, kl_ng = 0.f;
  for (int d = 0; d < DC; ++d) {
    float nt  = nat_tau[k * DC + d];
    float tau = nt * invc;
    float Bd  = nat_B[k * DC + d] - nt * nt * invc;   // recovered B
    float w   = n / Bd;
    float wt  = w * tau;
    sum_loglam += dg_a + LN2_F - logf(Bd);            // digamma(n/2)+ln2-lnB
    sum_wtau2  += wt * tau;

    float b = 0.5f * Bd, b0 = 0.5f * B0[k * DC + d];
    float kg = (a - a0) * dg_a - lg_a + lg_a0 + a0 * (logf(b) - logf(b0))
             + a * (b0 - b) / b;
    float dtau = tau - tau0[k * DC + d];
    float kn = 0.5f * (logf(c / c0k) + c0k * invc - 1.f + c0k * w * dtau * dtau);
    kl_ng += kg + kn;

    ws[WS_BPREP + bprep_idx(d, k)]      = -0.5f * w;  // multiplies x^2
    ws[WS_BPREP + bprep_idx(DC + d, k)] = wt;         // multiplies x
  }
  ws[WS_CONST + k] = e_log_pi
      + 0.5f * (sum_loglam - (float)DC * LOG_2PI_F - (float)DC * invc)
      - 0.5f * sum_wtau2;

  float klb = lnbetaf_(u0[k], v0[k]) - lnbetaf_(u, v)
            + (u - u0[k]) * dg_u + (v - v0[k]) * dg_v
            - (u + v - u0[k] - v0[k]) * dg_uv;
  s_kl[k] = klb + kl_ng;
  __syncthreads();
  for (int s = 64; s > 0; s >>= 1) {
    if (k < s) s_kl[k] += s_kl[k + s];
    __syncthreads();
  }
  if (k == 0) ws[WS_KL] = s_kl[0];
}

// 8 waves per block, 16 rows per wave -> 128 rows per block.
// B (64 KB, pre-swizzled into WMMA fragment order) is DMA'd LDS-resident once per
// block by the Tensor Data Mover, then the GEMM loop feeds from ds_load_b64.
// Each wave: logits(16x128) = [x^2 | x](16x128) @ B(128x128) via 256x V_WMMA_F32_16X16X4_F32
__global__ __launch_bounds__(256) void dpmm_main(
    const float* __restrict__ x, const float* __restrict__ ws,
    float* __restrict__ r_out, float* __restrict__ bsum) {
  __shared__ float sB[16384];                     // 64 KB: B fragments, reused for lse partials

  const float* __restrict__ cst = ws + WS_CONST;

  const int lane = threadIdx.x & 31;
  const int wid  = threadIdx.x >> 5;
  const int lo   = lane & 15;
  const int hi   = lane >> 4;
  const int rowBase = blockIdx.x * 128 + wid * 16;

  // ---- TDM: stage B into LDS (one copy per workgroup, issued by wave 0) ----
  if (wid == 0) {
    unsigned lds_off = (unsigned)(uintptr_t)&sB[0];          // flat addr low 32 bits == LDS offset
    unsigned long long ga = (unsigned long long)(uintptr_t)(ws + WS_BPREP);
    // D# group 0: count=1 | lds_addr | global_addr[56:0] | type=2
    u32x4 g0;
    g0[0] = 1u;                                              // count=1, gather off
    g0[1] = lds_off;                                         // lds_addr
    g0[2] = (unsigned)(ga & 0xffffffffu);                    // global_addr[31:0]
    g0[3] = (unsigned)((ga >> 32) & 0x01ffffffu) | (2u << 30); // global_addr[56:32] | type=2
    // D# group 1: data_size=4B, 1-D tensor of 16384 elements, tile = whole tensor
    i32x8 g1;
    g1[0] = (int)(2u << 16);           // workgroup_mask=0, data_size=2 (4 bytes)
    g1[1] = (int)(16384u << 16);       // tensor_dim0[15:0] in bits[31:16]
    g1[2] = (int)(1u << 16);           // tensor_dim0[31:16]=0, tensor_dim1=1
    g1[3] = (int)(16384u << 16);       // tensor_dim1[31:16]=0, tile_dim0=16384
    g1[4] = 0;                         // tile_dim1=0 (unused), tile_dim2=0
    g1[5] = 16384;                     // tensor_dim0_stride[31:0]
    g1[6] = 0;                         // stride hi, tensor_dim1_stride lo
    g1[7] = 0;
    i32x4 z4 = {0, 0, 0, 0};
    i32x8 z8 = {0, 0, 0, 0, 0, 0, 0, 0};
    __builtin_amdgcn_tensor_load_to_lds(g0, g1, z4, z4, z8, 0);
    __builtin_amdgcn_s_wait_tensorcnt(0);
  }
  __syncthreads();

  // ---- A-fragment source data (streaming, non-temporal): row rowBase+lo, d-pairs {4g+2*hi,+1}
  v2f ap[16];
  const float* xr = x + (size_t)(rowBase + lo) * DC;
#pragma unroll
  for (int g = 0; g < 16; ++g) {
    ap[g] = __builtin_nontemporal_load((const v2f*)(xr + 4 * g + 2 * hi));
  }

  v8f C[8];
#pragma unroll
  for (int t = 0; t < 8; ++t) C[t] = (v8f){0.f, 0.f, 0.f, 0.f, 0.f, 0.f, 0.f, 0.f};

#pragma unroll
  for (int kk = 0; kk < 32; ++kk) {
    v2f p = ap[kk & 15];
    v2f A;
    if (kk < 16) { A[0] = p[0] * p[0]; A[1] = p[1] * p[1]; }  // x^2 half of inner dim
    else         { A[0] = p[0];        A[1] = p[1];        }  // x half
#pragma unroll
    for (int t = 0; t < 8; ++t) {
      v2f B = *(const v2f*)&sB[((kk * 8 + t) * 32 + lane) * 2];  // ds_load_b64
      C[t] = __builtin_amdgcn_wmma_f32_16x16x4_f32(
          false, A, false, B, (short)0, C[t], false, false);
    }
  }

  // C layout: lane holds k = 16t+lo, rows j+8*hi (j = VGPR index 0..7)
#pragma unroll
  for (int t = 0; t < 8; ++t) {
    float ck = cst[16 * t + lo];
#pragma unroll
    for (int j = 0; j < 8; ++j) C[t][j] += ck;
  }

  float lse_acc = 0.f;
#pragma unroll
  for (int j = 0; j < 8; ++j) {
    float mx = -INFINITY;
#pragma unroll
    for (int t = 0; t < 8; ++t) mx = fmaxf(mx, C[t][j]);
    for (int s = 1; s < 16; s <<= 1) mx = fmaxf(mx, __shfl_xor(mx, s, 32));
    float sum = 0.f;
#pragma unroll
    for (int t = 0; t < 8; ++t) sum += expf(C[t][j] - mx);
    for (int s = 1; s < 16; s <<= 1) sum += __shfl_xor(sum, s, 32);
    float lse = mx + logf(sum);

    int row = rowBase + j + 8 * hi;
    float* orow = r_out + (size_t)row * KC;
#pragma unroll
    for (int t = 0; t < 8; ++t) {
      __builtin_nontemporal_store(expf(C[t][j] - lse), &orow[16 * t + lo]);
    }
    lse_acc += lse;
  }

  // deterministic per-block partial of sum(lse); reuse sB (B no longer needed)
  __syncthreads();
  if (lo == 0) sB[wid * 2 + hi] = lse_acc;
  __syncthreads();
  if (threadIdx.x == 0) {
    float s = 0.f;
    for (int i = 0; i < 16; ++i) s += sB[i];
    bsum[blockIdx.x] = s;
  }
}

__global__ __launch_bounds__(256) void dpmm_final(
    const float* __restrict__ bsumv, const float* __restrict__ klv,
    float* __restrict__ out_scalar, int nblocks) {
  __shared__ float s[256];
  float acc = 0.f;
  for (int i = threadIdx.x; i < nblocks; i += 256) acc += bsumv[i];
  s[threadIdx.x] = acc;
  __syncthreads();
  for (int st = 128; st > 0; st >>= 1) {
    if ((int)threadIdx.x < st) s[threadIdx.x] += s[threadIdx.x + st];
    __syncthreads();
  }
  if (threadIdx.x == 0) *out_scalar = *klv - s[0];  // -elbo = KL - data_elbo
}

extern "C" void kernel_launch(void* const* d_in, const int* in_sizes, int n_in,
                              void* d_out, int out_size, void* d_ws, size_t ws_size,
                              hipStream_t stream) {
  const float* x       = (const float*)d_in[0];
  const float* nat_u   = (const float*)d_in[1];
  const float* nat_v   = (const float*)d_in[2];
  const float* nat_tau = (const float*)d_in[3];
  const float* nat_c   = (const float*)d_in[4];
  const float* nat_n   = (const float*)d_in[5];
  const float* nat_B   = (const float*)d_in[6];
  const float* u0      = (const float*)d_in[7];
  const float* v0      = (const float*)d_in[8];
  const float* tau0    = (const float*)d_in[9];
  const float* c0      = (const float*)d_in[10];
  const float* n0      = (const float*)d_in[11];
  const float* B0      = (const float*)d_in[12];

  float* ws    = (float*)d_ws;
  float* r_out = (float*)d_out;
  float* neg_elbo = r_out + (size_t)NC * KC;

  dpmm_prep<<<1, 128, 0, stream>>>(nat_u, nat_v, nat_tau, nat_c, nat_n, nat_B,
                                   u0, v0, tau0, c0, n0, B0, ws);
  int nblocks = NC / 128;  // 1024
  dpmm_main<<<nblocks, 256, 0, stream>>>(x, ws, r_out, ws + WS_BSUM);
  dpmm_final<<<1, 256, 0, stream>>>(ws + WS_BSUM, ws + WS_KL, neg_elbo, nblocks);
}